// NeonKF_62646392980132
// MI455X (gfx1250) — compile-verified
//
#include <hip/hip_runtime.h>
#include <stdint.h>

// ---------------- problem constants (match reference) ----------------
#define B_ROWS  4096
#define T_TOT   4096
#define L_HIST  3072
#define FC_LEN  (T_TOT - L_HIST)     // 1024 forecast steps
#define CHUNK   64                   // timesteps per staged tile
#define RPB     32                   // rows per block == one wave32
#define PITCH   65                   // LDS row pitch in dwords (bank-conflict free)
#define NREG    (RPB * PITCH)        // dwords per (array, buffer) region = 2080
#define NCHUNK  (T_TOT / CHUNK)      // 64
#define HCHUNK  (L_HIST / CHUNK)     // 48 (history chunks)

typedef unsigned int v4u __attribute__((ext_vector_type(4)));
typedef int          v8i __attribute__((ext_vector_type(8)));
typedef int          v4i __attribute__((ext_vector_type(4)));

#if defined(__HIP_DEVICE_COMPILE__) && __has_builtin(__builtin_amdgcn_tensor_load_to_lds)
#define KF_USE_TDM 1
#else
#define KF_USE_TDM 0
#endif

#if KF_USE_TDM
// Issue one TDM 2-D tile load: RPB rows x CHUNK f32, row stride T_TOT elements,
// landing in LDS at lds_byte_addr with 1-dword pad after every 64 dwords
// (=> 65-dword LDS row pitch, bank-conflict-free column reads).
__device__ __forceinline__ void kf_tdm_load_tile(uint32_t lds_byte_addr, const void* gptr)
{
    const uint64_t ga = (uint64_t)(uintptr_t)gptr;
    v4u g0;
    g0[0] = 1u;                                                  // count=1 (valid), no gather
    g0[1] = lds_byte_addr;                                       // lds_addr (bytes)
    g0[2] = (uint32_t)ga;                                        // global_addr[31:0]
    g0[3] = ((uint32_t)(ga >> 32) & 0x01FFFFFFu) | (2u << 30);   // global_addr[56:32] | type=2
    v8i g1;
    g1[0] = (int)((2u << 16) | (1u << 20) | (5u << 22));         // data_size=4B, pad_en, interval=64dw, pad=1dw
    g1[1] = (int)((uint32_t)CHUNK << 16);                        // tensor_dim0 = 64 (low16 in [31:16])
    g1[2] = (int)((uint32_t)RPB   << 16);                        // tensor_dim1 = 32
    g1[3] = (int)((uint32_t)CHUNK << 16);                        // tile_dim0   = 64
    g1[4] = (int)RPB;                                            // tile_dim1 = 32, tile_dim2 = 0
    g1[5] = (int)T_TOT;                                          // tensor_dim0_stride = 4096 elements
    g1[6] = 0;                                                   // stride0 hi / stride1 lo
    g1[7] = 0;
    v4i gz4 = {0, 0, 0, 0};                                      // 2-D tile: groups 2/3 unused
    v8i gz8 = {0, 0, 0, 0, 0, 0, 0, 0};
    __builtin_amdgcn_tensor_load_to_lds(g0, g1, gz4, gz4, gz8, 0);
}
#endif

// One 64-step chunk of the Kalman scan. Pointers are per-lane row bases with
// unit stride in time. Replicates reference arithmetic order exactly.
__device__ __forceinline__ void kf_compute_chunk(
    int c, int b,
    const float* __restrict__ To, const float* __restrict__ Tap,
    const float* __restrict__ dtp,
    float& s, float& P, float& Ta_carry,
    float k, float q, float R,
    float* __restrict__ outP, float* __restrict__ outV)
{
    if (c < HCHUNK) {
        const int i0 = (c == 0) ? 1 : 0;   // global step n=0 has no work
        for (int i = i0; i < CHUNK; ++i) {
            const float dtt = fmaxf(dtp[i], 1.0f);
            const float Tav = (i == 0) ? Ta_carry : Tap[i - 1];
            // predict
            s = s - k * (s - Tav) * dtt;
            const float Fs = 1.0f - k * dtt;
            P = fminf(fmaxf(Fs * Fs * P + q * dtt, 1e-10f), 1.0e6f);
            // update
            const float S = P + R;
            const float K = P / S;
            const float y = To[i];
            s = s + K * (y - s);
            const float omk = 1.0f - K;
            P = omk * omk * P + K * K * R;
        }
    } else {
        const int jbase = c * CHUNK - L_HIST;
        for (int i = 0; i < CHUNK; ++i) {
            const float dtt = fmaxf(dtp[i], 1.0f);
            const float Tav = (i == 0) ? Ta_carry : Tap[i - 1];
            s = s - k * (s - Tav) * dtt;
            const float Fs = 1.0f - k * dtt;
            P = fminf(fmaxf(Fs * Fs * P + q * dtt, 1e-10f), 1.0e6f);
            const size_t o = (size_t)b * FC_LEN + (size_t)(jbase + i);
            outP[o] = s;
            outV[o] = P;
        }
    }
    Ta_carry = Tap[CHUNK - 1];
}

__global__ __launch_bounds__(RPB) void NeonKF_kernel(
    const float* __restrict__ T_obs, const float* __restrict__ T_air,
    const float* __restrict__ dt,
    const float* __restrict__ k_raw_p, const float* __restrict__ log_q_p,
    const float* __restrict__ log_r_p, const float* __restrict__ log_p0_p,
    float* __restrict__ out)
{
    __shared__ float smem[3 * 2 * NREG];  // {T_obs,T_air,dt} x double-buffer, 65-dw pitch

    const int r    = threadIdx.x;             // lane -> row within tile
    const int row0 = blockIdx.x * RPB;
    const int b    = row0 + r;

    const float k  = log1pf(expf(k_raw_p[0]));   // softplus
    const float q  = expf(log_q_p[0]);
    const float R  = expf(log_r_p[0]);

    float s = T_obs[(size_t)b * T_TOT];          // s0 = T_obs[:,0]
    float P = expf(log_p0_p[0]);                 // P0
    float Ta_carry = 0.0f;                       // only read from chunk 1 onward

    float* outP = out;
    float* outV = out + (size_t)B_ROWS * FC_LEN;

#if KF_USE_TDM
    const uint32_t lds0 = (uint32_t)(uintptr_t)(&smem[0]);  // LDS aperture: low 32 bits = LDS offset
    const size_t rowbase = (size_t)row0 * T_TOT;

    // arr: 0=T_obs, 1=T_air, 2=dt ; region byte offset = (arr*2+buf)*NREG*4
    auto issue_chunk = [&](int c, int buf, bool with_obs) {
        const size_t goff = (rowbase + (size_t)c * CHUNK) * sizeof(float);
        if (with_obs)
            kf_tdm_load_tile(lds0 + (uint32_t)((0 * 2 + buf) * NREG * 4),
                             (const char*)T_obs + goff);
        kf_tdm_load_tile(lds0 + (uint32_t)((1 * 2 + buf) * NREG * 4),
                         (const char*)T_air + goff);
        kf_tdm_load_tile(lds0 + (uint32_t)((2 * 2 + buf) * NREG * 4),
                         (const char*)dt + goff);
    };

    issue_chunk(0, 0, true);                                  // prime the pipe (3 ops pending)
    for (int c = 0; c < NCHUNK; ++c) {
        const int buf = c & 1;
        if (c + 1 < NCHUNK)
            issue_chunk(c + 1, (c + 1) & 1, (c + 1) < HCHUNK);
        // In-order TENSORcnt: wait until only the *next* chunk's ops remain.
        if (c + 1 >= NCHUNK)      __builtin_amdgcn_s_wait_tensorcnt(0);
        else if (c + 1 < HCHUNK)  __builtin_amdgcn_s_wait_tensorcnt(3);
        else                      __builtin_amdgcn_s_wait_tensorcnt(2);
        asm volatile("" ::: "memory");   // TDM wrote LDS behind the compiler's back

        const float* To  = &smem[(0 * 2 + buf) * NREG + r * PITCH];
        const float* Tap = &smem[(1 * 2 + buf) * NREG + r * PITCH];
        const float* dtp = &smem[(2 * 2 + buf) * NREG + r * PITCH];
        kf_compute_chunk(c, b, To, Tap, dtp, s, P, Ta_carry, k, q, R, outP, outV);
    }
#else
    // Fallback (host pass / builtin unavailable): read straight from global.
    for (int c = 0; c < NCHUNK; ++c) {
        const size_t base = (size_t)b * T_TOT + (size_t)c * CHUNK;
        kf_compute_chunk(c, b, T_obs + base, T_air + base, dt + base,
                         s, P, Ta_carry, k, q, R, outP, outV);
    }
#endif
}

extern "C" void kernel_launch(void* const* d_in, const int* in_sizes, int n_in,
                              void* d_out, int out_size, void* d_ws, size_t ws_size,
                              hipStream_t stream)
{
    (void)in_sizes; (void)n_in; (void)d_ws; (void)ws_size; (void)out_size;
    // setup_inputs order: 0=T_obs 1=T_air 2=wind 3=par 4=dt 5=k_raw 6=log_q 7=log_r 8=log_p0 9=L_hist
    const float* T_obs = (const float*)d_in[0];
    const float* T_air = (const float*)d_in[1];
    const float* dt    = (const float*)d_in[4];
    const float* k_raw = (const float*)d_in[5];
    const float* log_q = (const float*)d_in[6];
    const float* log_r = (const float*)d_in[7];
    const float* log_p0= (const float*)d_in[8];
    float* out = (float*)d_out;

    dim3 grid(B_ROWS / RPB);   // 128 blocks
    dim3 block(RPB);           // 32 threads = 1 wave32
    NeonKF_kernel<<<grid, block, 0, stream>>>(T_obs, T_air, dt,
                                              k_raw, log_q, log_r, log_p0, out);
}